// GaussianTransformLayer_86955907874878
// MI455X (gfx1250) — compile-verified
//
#include <hip/hip_runtime.h>

// CDNA5 / gfx1250 Haar-DWT along S expressed as D = M(16x16) x X(16x16)
// via chained V_WMMA_F32_16X16X4_F32 (native f32 matrix pipe, exact f32 math).
// Memory-bound kernel (512 MiB streaming, ~23us floor @ 23.3 TB/s) -> NT hints.

typedef __attribute__((ext_vector_type(2))) float v2f;
typedef __attribute__((ext_vector_type(8))) float v8f;

__global__ __launch_bounds__(256) void haar_dwt_wmma_kernel(
        const float* __restrict__ x, float* __restrict__ out) {
    constexpr int S    = 4096;
    constexpr int C    = 512;
    constexpr int HALF = S / 2; // 2048
    const float r = 0.70710678118654752440f; // 1/sqrt(2)

    const int tid  = threadIdx.x;
    const int wave = tid >> 5;     // 8 waves per block
    const int lane = tid & 31;

    const int b = blockIdx.x >> 8;   // 256 seq-tiles of 16 rows per batch
    const int t = blockIdx.x & 255;  // seq tile: input rows 16t .. 16t+15

    const bool lo  = (lane < 16);
    const int  m   = lane & 15;      // A-matrix row / B,N column index
    const int  col = lane & 15;
    const int  hi  = lo ? 0 : 1;

    // --- A matrix (constant per wave), WMMA 16x4 f32 layout:
    //     VGPR0: K = 4k + (lo ? 0 : 2)  (even K -> tap is always +r on match)
    //     VGPR1: K = 4k + (lo ? 1 : 3)  (odd  K -> tap is +r (cA rows) / -r (cD rows))
    // Haar matrix M[m,s] nonzero iff (s>>1) == (m&7); odd-s tap negated for m>=8.
    const float sgn = (m & 8) ? -r : r;
    const int   m7  = m & 7;
    v2f A[4];
#pragma unroll
    for (int k = 0; k < 4; ++k) {
        const bool p = (m7 == 2 * k + hi);   // single predicate per k-chunk
        A[k].x = p ? r   : 0.0f;
        A[k].y = p ? sgn : 0.0f;
    }

    // B-matrix row offsets per lane half (4x16 f32 layout mirrors A):
    const int r0 = lo ? 0 : 2;
    const int r1 = lo ? 1 : 3;

    const size_t xbase = ((size_t)b * S + (size_t)t * 16) * C;

    // Per-lane output row base:
    //   lanes 0..15  -> D rows 0..7  = cA pairs 8t+v  -> out row (8t + v)
    //   lanes 16..31 -> D rows 8..15 = cD pairs 8t+v  -> out row (HALF + 8t + v)
    const size_t obase = (size_t)b * S * C
                       + (size_t)(lo ? (8 * t) : (HALF + 8 * t)) * C
                       + (size_t)col;

#pragma unroll
    for (int ct = 0; ct < 4; ++ct) {
        const int c0 = (wave * 4 + ct) * 16;   // 8 waves x 4 tiles = 32 channel tiles
        const float* xp = x + xbase + (size_t)c0 + (size_t)col;

        v8f acc = {};
#pragma unroll
        for (int k = 0; k < 4; ++k) {
            v2f Bm;
            Bm.x = __builtin_nontemporal_load(xp + (size_t)(4 * k + r0) * C);
            Bm.y = __builtin_nontemporal_load(xp + (size_t)(4 * k + r1) * C);
            // D = A_k * B_k + acc   (v_wmma_f32_16x16x4_f32, chained accumulation)
            acc = __builtin_amdgcn_wmma_f32_16x16x4_f32(
                      /*neg_a=*/false, A[k],
                      /*neg_b=*/false, Bm,
                      /*c_mod=*/(short)0, acc,
                      /*reuse_a=*/false, /*reuse_b=*/false);
        }

        float* op = out + obase + (size_t)c0;
#pragma unroll
        for (int v = 0; v < 8; ++v) {
            __builtin_nontemporal_store(acc[v], op + (size_t)v * C);
        }
    }
}

extern "C" void kernel_launch(void* const* d_in, const int* in_sizes, int n_in,
                              void* d_out, int out_size, void* d_ws, size_t ws_size,
                              hipStream_t stream) {
    (void)in_sizes; (void)n_in; (void)out_size; (void)d_ws; (void)ws_size;

    const float* x = (const float*)d_in[0];
    float* out     = (float*)d_out;

    constexpr int B = 32;
    constexpr int S = 4096;
    const int grid = B * (S / 16);   // 8192 blocks, one 16-row seq tile each

    haar_dwt_wmma_kernel<<<grid, 256, 0, stream>>>(x, out);
}